// DecoderRNN_70918499992252
// MI455X (gfx1250) — compile-verified
//
#include <hip/hip_runtime.h>
#include <math.h>

#define EMB   1024
#define VOCAB 50257

typedef float v2f __attribute__((ext_vector_type(2)));
typedef float v8f __attribute__((ext_vector_type(8)));

// ---------------------------------------------------------------------------
// K1: GRU gate GEMVs via V_WMMA_F32_16X16X4_F32.
// Blocks 0..23 compute gi = W_ih @ relu(emb[token]) + b_ih   (3072 rows)
// Blocks 24..47 compute gh = W_hh @ h               + b_hh   (3072 rows)
// Each wave owns a 16-row tile; split-K (two 512-wide halves) for ILP.
// A operand: 16x4 weight tile (float2/lane, ISA 32-bit A layout).
// B operand: vector chunk broadcast over all 16 columns -> D[m][*] = row dot.
// ---------------------------------------------------------------------------
__global__ __launch_bounds__(256) void gru_gates_kernel(
    const int* __restrict__ token, const float* __restrict__ hidden,
    const float* __restrict__ emb,
    const float* __restrict__ W_ih, const float* __restrict__ W_hh,
    const float* __restrict__ b_ih, const float* __restrict__ b_hh,
    float* __restrict__ gi, float* __restrict__ gh)
{
    __shared__ float sv[EMB];
    const int tid = threadIdx.x;
    const bool isIH = (blockIdx.x < 24);

    if (isIH) {
        const float* er = emb + (size_t)token[0] * EMB;
        for (int i = tid; i < EMB; i += 256) {
            float v = er[i];
            sv[i] = v > 0.f ? v : 0.f;           // embedding + ReLU
        }
    } else {
        for (int i = tid; i < EMB; i += 256) sv[i] = hidden[i];
    }
    __syncthreads();

    const int wave = tid >> 5;
    const int lane = tid & 31;
    const int m    = lane & 15;
    const int koff = (lane >> 4) << 1;           // 0 for lanes 0-15, 2 for 16-31

    const float* Wm   = isIH ? W_ih : W_hh;
    const float* bias = isIH ? b_ih : b_hh;
    float*       outp = isIH ? gi   : gh;
    const int tileInMat = (isIH ? blockIdx.x : blockIdx.x - 24) * 8 + wave; // 0..191
    const int row0 = tileInMat * 16;

    const float* wrow = Wm + (size_t)(row0 + m) * EMB + koff;

    v8f c0 = {}, c1 = {};
    #pragma unroll 4
    for (int base = 0; base < EMB / 2; base += 4) {
        v2f a0 = *(const v2f*)(wrow + base);
        v2f a1 = *(const v2f*)(wrow + base + EMB / 2);
        v2f b0, b1;
        b0.x = sv[base + koff];
        b0.y = sv[base + koff + 1];
        b1.x = sv[base + koff + EMB / 2];
        b1.y = sv[base + koff + 1 + EMB / 2];
        c0 = __builtin_amdgcn_wmma_f32_16x16x4_f32(false, a0, false, b0,
                                                   (short)0, c0, false, false);
        c1 = __builtin_amdgcn_wmma_f32_16x16x4_f32(false, a1, false, b1,
                                                   (short)0, c1, false, false);
    }
    v8f c = c0 + c1;

    // D[m][n] identical over n: lane 0 holds rows 0..7 (N=0), lane 16 rows 8..15.
    if (lane == 0) {
        #pragma unroll
        for (int mm = 0; mm < 8; ++mm)
            outp[row0 + mm] = c[mm] + bias[row0 + mm];
    } else if (lane == 16) {
        #pragma unroll
        for (int mm = 0; mm < 8; ++mm)
            outp[row0 + 8 + mm] = c[mm] + bias[row0 + 8 + mm];
    }
}

// ---------------------------------------------------------------------------
// K2: gate nonlinearities + h_new (PyTorch r,z,n order).
// ---------------------------------------------------------------------------
__global__ __launch_bounds__(256) void gru_combine_kernel(
    const float* __restrict__ gi, const float* __restrict__ gh,
    const float* __restrict__ hidden,
    float* __restrict__ hnew, float* __restrict__ out_h)
{
    int j = blockIdx.x * 256 + threadIdx.x;
    if (j >= EMB) return;
    float r = 1.f / (1.f + expf(-(gi[j]         + gh[j])));
    float z = 1.f / (1.f + expf(-(gi[EMB + j]   + gh[EMB + j])));
    float n = tanhf(gi[2 * EMB + j] + r * gh[2 * EMB + j]);
    float h = hidden[j];
    float v = (1.f - z) * n + z * h;
    hnew[j]  = v;
    out_h[j] = v;
}

// ---------------------------------------------------------------------------
// K3: logits = W_out @ h_new + b_out (one vocab row per wave, b128 streams),
// plus per-block online-softmax partials (max, sum-exp).
// Logits are written straight into d_out and normalized in-place later.
// ---------------------------------------------------------------------------
__global__ __launch_bounds__(256) void logits_kernel(
    const float* __restrict__ Wout, const float* __restrict__ bout,
    const float* __restrict__ hnew,
    float* __restrict__ logits, float2* __restrict__ partials)
{
    __shared__ float shh[EMB];
    __shared__ float wl[8];
    const int tid = threadIdx.x;
    for (int i = tid; i < EMB; i += 256) shh[i] = hnew[i];
    __syncthreads();

    const int wave = tid >> 5, lane = tid & 31;
    const int v = blockIdx.x * 8 + wave;
    float logit = -INFINITY;
    if (v < VOCAB) {
        const float* wr = Wout + (size_t)v * EMB;
        float acc = 0.f;
        #pragma unroll
        for (int i = 0; i < 8; ++i) {
            int col = i * 128 + lane * 4;        // 512B contiguous per wave-load
            float4 w = *(const float4*)(wr + col);
            float4 x = *(const float4*)(&shh[col]);
            acc += w.x * x.x + w.y * x.y + w.z * x.z + w.w * x.w;
        }
        #pragma unroll
        for (int off = 16; off; off >>= 1)
            acc += __shfl_xor(acc, off, 32);
        if (lane == 0) {
            logit = acc + bout[v];
            logits[v] = logit;
        }
    }
    if (lane == 0) wl[wave] = logit;
    __syncthreads();
    if (tid == 0) {
        float mb = -INFINITY;
        #pragma unroll
        for (int w = 0; w < 8; ++w) mb = fmaxf(mb, wl[w]);
        float sb = 0.f;
        #pragma unroll
        for (int w = 0; w < 8; ++w) sb += expf(wl[w] - mb);  // expf(-inf)=0
        partials[blockIdx.x] = make_float2(mb, sb);
    }
}

// ---------------------------------------------------------------------------
// K4: reduce (max, sumexp) partials -> logZ (single block).
// ---------------------------------------------------------------------------
__global__ __launch_bounds__(256) void reduce_kernel(
    const float2* __restrict__ partials, int P, float* __restrict__ logZ)
{
    __shared__ float sm[256], ss[256];
    const int tid = threadIdx.x;
    float m = -INFINITY, s = 0.f;
    for (int i = tid; i < P; i += 256) {
        float2 p = partials[i];
        if (p.x > m) { s = s * expf(m - p.x) + p.y; m = p.x; }
        else if (p.x != -INFINITY) { s += p.y * expf(p.x - m); }
    }
    sm[tid] = m; ss[tid] = s;
    __syncthreads();
    for (int str = 128; str; str >>= 1) {
        if (tid < str) {
            float m1 = sm[tid], s1 = ss[tid];
            float m2 = sm[tid + str], s2 = ss[tid + str];
            if (m2 > m1)                 { s1 = s1 * expf(m1 - m2) + s2; m1 = m2; }
            else if (m2 != -INFINITY)    { s1 += s2 * expf(m2 - m1); }
            sm[tid] = m1; ss[tid] = s1;
        }
        __syncthreads();
    }
    if (tid == 0) logZ[0] = sm[0] + logf(ss[0]);
}

// ---------------------------------------------------------------------------
// K5: log_probs = logits - logZ (in place on d_out).
// ---------------------------------------------------------------------------
__global__ __launch_bounds__(256) void finalize_kernel(
    float* __restrict__ out_logits, const float* __restrict__ logZ)
{
    int v = blockIdx.x * 256 + threadIdx.x;
    if (v < VOCAB) out_logits[v] -= logZ[0];
}

// ---------------------------------------------------------------------------
extern "C" void kernel_launch(void* const* d_in, const int* in_sizes, int n_in,
                              void* d_out, int out_size, void* d_ws, size_t ws_size,
                              hipStream_t stream)
{
    (void)in_sizes; (void)n_in; (void)out_size; (void)ws_size;
    const int*   token  = (const int*)  d_in[0];
    const float* hidden = (const float*)d_in[1];
    const float* emb    = (const float*)d_in[2];
    const float* W_ih   = (const float*)d_in[3];
    const float* W_hh   = (const float*)d_in[4];
    const float* b_ih   = (const float*)d_in[5];
    const float* b_hh   = (const float*)d_in[6];
    const float* W_out  = (const float*)d_in[7];
    const float* b_out  = (const float*)d_in[8];

    float* out = (float*)d_out;              // [0,VOCAB): log_probs, then h_new
    float* ws  = (float*)d_ws;
    float*  gi       = ws;                   // 3072
    float*  gh       = ws + 3072;            // 3072
    float*  hnew     = ws + 6144;            // 1024
    float2* partials = (float2*)(ws + 7168); // 6283 float2 (8B aligned)
    float*  logZ     = ws + 7168 + 2 * 6283;

    const int NB = (VOCAB + 7) / 8;          // 6283 logit blocks

    gru_gates_kernel  <<<48, 256, 0, stream>>>(token, hidden, emb, W_ih, W_hh,
                                               b_ih, b_hh, gi, gh);
    gru_combine_kernel<<<(EMB + 255) / 256, 256, 0, stream>>>(gi, gh, hidden,
                                                              hnew, out + VOCAB);
    logits_kernel     <<<NB, 256, 0, stream>>>(W_out, b_out, hnew, out, partials);
    reduce_kernel     <<<1, 256, 0, stream>>>(partials, NB, logZ);
    finalize_kernel   <<<(VOCAB + 255) / 256, 256, 0, stream>>>(out, logZ);
}